// DisparityWarping_45595372814894
// MI455X (gfx1250) — compile-verified
//
#include <hip/hip_runtime.h>
#include <hip/hip_bf16.h>

// Problem shape (fixed by setup_inputs):
//   x:         [4, 16, 320, 1024] float32
//   disparity: [4,  1, 320, 1024] float32 in [0, 64)
// Outputs (concatenated flat in d_out, float32):
//   warped:    [4, 16, 320, 1024]
//   mask:      [4,  1, 320, 1024]  (bool -> 0.0f / 1.0f)

#define DW_N 4
#define DW_C 16
#define DW_H 320
#define DW_W 1024

#if defined(__HIP_DEVICE_COMPILE__) && defined(__gfx1250__)
#define DW_GFX1250 1
#endif

__global__ __launch_bounds__(DW_W) void DisparityWarping_45595372814894_kernel(
    const float* __restrict__ x,
    const float* __restrict__ disp,
    float* __restrict__ out_img,
    float* __restrict__ out_mask)
{
    __shared__ int   s_gjnew[DW_W];   // scatter-max buffer (one row)
    __shared__ float s_disp[DW_W];    // staged disparity row

    const int j   = (int)threadIdx.x;        // column 0..W-1
    const int row = (int)blockIdx.x;         // 0..N*H-1
    const int b   = row / DW_H;
    const int i   = row - b * DW_H;
    const int HW  = DW_H * DW_W;

    const float* drow = disp + (size_t)row * DW_W;
    const float* px0  = x    + (size_t)b * DW_C * HW;   // x[b, 0, 0, 0]

    // ---- Stage the disparity row into LDS.
    // On gfx1250 use the async global->LDS path (ASYNCcnt-tracked) if the
    // toolchain exposes the builtin; otherwise plain load+store.
    // Probe result (round 1): builtin exists; param 0 is generic 'int *'.
#if defined(DW_GFX1250) && __has_builtin(__builtin_amdgcn_global_load_async_to_lds_b32)
    __builtin_amdgcn_global_load_async_to_lds_b32(
        (int*)(drow + j),
        (int*)&s_disp[j],
        /*offset=*/0, /*cpol=*/0);
#else
    s_disp[j] = drow[j];
#endif

    // Init scatter-max buffer (occlusion): reference buffer starts at 0.
    s_gjnew[j] = 0;

#if defined(DW_GFX1250) && __has_builtin(__builtin_amdgcn_global_load_async_to_lds_b32)
#if __has_builtin(__builtin_amdgcn_s_wait_asynccnt)
    __builtin_amdgcn_s_wait_asynccnt(0);
#else
    asm volatile("s_wait_asynccnt 0" ::: "memory");
#endif
#endif
    __syncthreads();   // s_gjnew init + s_disp visible to the whole block

    const float d = s_disp[j];

    // ---- Occlusion scatter-max: trans[j] = max(j - int(d), 0);
    // last-write-wins over ordered j == scatter-max of j.
    int di    = (int)d;                 // trunc toward 0 (d >= 0 -> floor)
    int trans = j - di;
    if (trans < 0) trans = 0;
    atomicMax(&s_gjnew[trans], j);      // ds_max on LDS

    // ---- Grid / bilinear setup, replicating the reference float math.
    const float jf = (float)j;
    const float gx = 2.0f * (jf - d) / (float)(DW_W - 1) - 1.0f;
    const float gy = 2.0f * (float)i / (float)(DW_H - 1) - 1.0f;
    const int   cnt = (fabsf(gx) <= 1.0f ? 1 : 0) + (fabsf(gy) <= 1.0f ? 1 : 0);

    const float gxp = (gx + 1.0f) * (float)(DW_W - 1) * 0.5f;
    const float gyp = (gy + 1.0f) * (float)(DW_H - 1) * 0.5f;

    const float x0f = floorf(gxp);
    const float y0f = floorf(gyp);
    const float wx1 = gxp - x0f, wx0 = 1.0f - wx1;
    const float wy1 = gyp - y0f, wy0 = 1.0f - wy1;
    const float x1f = x0f + 1.0f, y1f = y0f + 1.0f;

    const float vx0 = (x0f >= 0.0f && x0f <= (float)(DW_W - 1)) ? 1.0f : 0.0f;
    const float vx1 = (x1f >= 0.0f && x1f <= (float)(DW_W - 1)) ? 1.0f : 0.0f;
    const float vy0 = (y0f >= 0.0f && y0f <= (float)(DW_H - 1)) ? 1.0f : 0.0f;
    const float vy1 = (y1f >= 0.0f && y1f <= (float)(DW_H - 1)) ? 1.0f : 0.0f;

    const float w00 = wx0 * wy0 * vx0 * vy0;
    const float w10 = wx1 * wy0 * vx1 * vy0;
    const float w01 = wx0 * wy1 * vx0 * vy1;
    const float w11 = wx1 * wy1 * vx1 * vy1;

    // Clipped integer gather indices (reference clips then gathers).
    const int xi0 = (int)fminf(fmaxf(x0f, 0.0f), (float)(DW_W - 1));
    const int xi1 = (int)fminf(fmaxf(x1f, 0.0f), (float)(DW_W - 1));
    const int yi0 = (int)fminf(fmaxf(y0f, 0.0f), (float)(DW_H - 1));
    const int yi1 = (int)fminf(fmaxf(y1f, 0.0f), (float)(DW_H - 1));

    const int a00 = yi0 * DW_W + xi0;
    const int a10 = yi0 * DW_W + xi1;
    const int a01 = yi1 * DW_W + xi0;
    const int a11 = yi1 * DW_W + xi1;

    // Warm the gather rows (global_prefetch_b8) while the DS scatter phase
    // completes across the block.
    __builtin_prefetch(px0 + a00, 0, 0);
    __builtin_prefetch(px0 + a11, 0, 0);
    __builtin_prefetch(px0 + 8 * HW + a00, 0, 0);
    __builtin_prefetch(px0 + 8 * HW + a11, 0, 0);

    __syncthreads();   // all atomicMax done

    const int gj_back = s_gjnew[trans];
    int diff = gj_back - j;
    if (diff < 0) diff = -diff;
    const int occ = (diff <= 1) ? 1 : 0;
    out_mask[(size_t)row * DW_W + j] = ((cnt + occ) == 3) ? 1.0f : 0.0f;

    // ---- 16-channel bilinear gather: coalesced loads (neighbors in j read
    // neighboring addresses) + coalesced stores. Sum order matches reference:
    // c(x0,y0) + c(x1,y0) + c(x0,y1) + c(x1,y1).
    float* po = out_img + (size_t)b * DW_C * HW + (size_t)i * DW_W + j;
#pragma unroll 4
    for (int c = 0; c < DW_C; ++c) {
        const float* p = px0 + (size_t)c * HW;
        float v = p[a00] * w00 + p[a10] * w10 + p[a01] * w01 + p[a11] * w11;
        po[(size_t)c * HW] = v;
    }
}

extern "C" void kernel_launch(void* const* d_in, const int* in_sizes, int n_in,
                              void* d_out, int out_size, void* d_ws, size_t ws_size,
                              hipStream_t stream) {
    (void)in_sizes; (void)n_in; (void)out_size; (void)d_ws; (void)ws_size;

    const float* x    = (const float*)d_in[0];
    const float* disp = (const float*)d_in[1];

    float* out_img  = (float*)d_out;
    float* out_mask = out_img + (size_t)DW_N * DW_C * DW_H * DW_W;

    dim3 grid(DW_N * DW_H);   // one block per image row
    dim3 block(DW_W);         // 1024 threads = 32 wave32 waves

    DisparityWarping_45595372814894_kernel<<<grid, block, 0, stream>>>(
        x, disp, out_img, out_mask);
}